// Lfm2ShortConv_73821897884222
// MI455X (gfx1250) — compile-verified
//
#include <hip/hip_runtime.h>
#include <hip/hip_bf16.h>

typedef __attribute__((ext_vector_type(2))) float v2f;
typedef __attribute__((ext_vector_type(8))) float v8f;

// Exact-fp32 CDNA5 matrix op: D(16x16,f32) = A(16x4,f32) x B(4x16,f32) + C.
// A layout (ISA 7.12.2): lanes 0-15 -> M=lane, K=0,1 in v[0],v[1]; lanes 16-31 -> M=lane-16, K=2,3.
// B layout (4x16):       lanes 0-15 -> N=lane, K=0,1;               lanes 16-31 -> N=lane-16, K=2,3.
// C/D layout: VGPR r, lanes 0-15 -> (M=r, N=lane); lanes 16-31 -> (M=r+8, N=lane-16).
__device__ __forceinline__ v8f wmma_f32_k4(v2f a, v2f b, v8f c) {
    return __builtin_amdgcn_wmma_f32_16x16x4_f32(
        /*neg_a=*/false, a, /*neg_b=*/false, b,
        /*c_mod=*/(short)0, c, /*reuse_a=*/false, /*reuse_b=*/false);
}

#define KC  64   // k-slab staged in LDS per block
#define LDA 68   // padded LDS row stride (floats): 272 B -> 16B-aligned rows,
                 // fragment reads hit banks (4*lane + c) % 64 -> conflict-free

// ---------------------------------------------------------------------------
// Kernel 1: bulk copy conv_state -> new_conv_state (float4 grid-stride).
// Rows hit by req_pool_indices are overwritten later by the fused kernel.
// ---------------------------------------------------------------------------
__global__ void __launch_bounds__(256)
lfm2_copy_state_kernel(const float* __restrict__ src,
                       float* __restrict__ dst,
                       size_t n /* floats */) {
    size_t n4 = n >> 2;
    size_t i = (size_t)blockIdx.x * blockDim.x + threadIdx.x;
    size_t stride = (size_t)gridDim.x * blockDim.x;
    const float4* s4 = (const float4*)src;
    float4* d4 = (float4*)dst;
    for (size_t k = i; k < n4; k += stride) d4[k] = s4[k];
    for (size_t k = (n4 << 2) + i; k < n; k += stride) dst[k] = src[k];  // tail
}

// ---------------------------------------------------------------------------
// Kernel 2: fused in_proj (3 chunks of w_in) + gating + depthwise conv +
// state scatter. Block = 8 waves sharing one 16-row A tile staged in LDS;
// each wave owns one 16-col tile and 3 accumulators (B/C/x chunks).
// ---------------------------------------------------------------------------
__global__ void __launch_bounds__(256)
lfm2_proj_conv_kernel(const float* __restrict__ hid,        // [B,H]
                      const float* __restrict__ w_in,       // [3H,H]
                      const float* __restrict__ conv_state, // [P,H,K-1]
                      const int*   __restrict__ idx,        // [B]
                      const float* __restrict__ conv_w,     // [H,K]
                      float* __restrict__ z,                // [B,H] (workspace)
                      float* __restrict__ out_state,        // [P,H,K-1]
                      int B, int H, int K) {
    __shared__ float As[16 * LDA];

    const int tid  = threadIdx.x;
    const int lane = tid & 31;
    const int wave = tid >> 5;
    const int hi16 = lane >> 4;
    const int l15  = lane & 15;
    const int i = blockIdx.y;            // 16-row tile over batch B
    const int j = blockIdx.x * 8 + wave; // 16-col tile over hidden H
    const size_t Hs = (size_t)H;

    // Cooperative A-slab loader: thread t loads row t/16, 4 cols at (t%16)*4.
    const int arow = tid >> 4;
    const int acol = (tid & 15) << 2;
    const float* Aglb = hid + (size_t)(i * 16 + arow) * Hs + acol;
    float4* Alds = (float4*)&As[arow * LDA + acol];

    // Streaming B fragment pointers (contiguous float2 per k-step).
    const float* BpB = w_in + ((size_t)(0 * H) + (size_t)(j * 16 + l15)) * Hs + 2 * hi16;
    const float* BpC = w_in + ((size_t)(1 * H) + (size_t)(j * 16 + l15)) * Hs + 2 * hi16;
    const float* BpX = w_in + ((size_t)(2 * H) + (size_t)(j * 16 + l15)) * Hs + 2 * hi16;

    v8f accB = {}, accC = {}, accX = {};
    for (int k0 = 0; k0 < H; k0 += KC) {
        __syncthreads();                         // previous slab fully consumed
        *Alds = *(const float4*)(Aglb + k0);     // stage A slab (global->LDS)
        __builtin_prefetch(BpB + k0 + KC, 0, 0); // pull next weight slab toward L2
        __builtin_prefetch(BpC + k0 + KC, 0, 0);
        __builtin_prefetch(BpX + k0 + KC, 0, 0);
        __syncthreads();
#pragma unroll
        for (int kk = 0; kk < KC; kk += 4) {
            v2f a  = *(const v2f*)&As[l15 * LDA + kk + 2 * hi16];  // ds_load_b64
            v2f bB = *(const v2f*)(BpB + k0 + kk);
            v2f bC = *(const v2f*)(BpC + k0 + kk);
            v2f bX = *(const v2f*)(BpX + k0 + kk);
            accB = wmma_f32_k4(a, bB, accB);
            accC = wmma_f32_k4(a, bC, accC);
            accX = wmma_f32_k4(a, bX, accX);
        }
    }

    v8f bx = accB * accX;   // element layouts match -> pure register product

    const int h = j * 16 + l15;        // hidden column owned by this lane
    const int Km1 = K - 1;
    float cw[8];                       // K <= 8 (K == 4 here)
    for (int t = 0; t < K && t < 8; ++t) cw[t] = conv_w[(size_t)h * K + t];

    for (int r = 0; r < 8; ++r) {
        const int b = i * 16 + r + 8 * hi16;   // batch row of this C/D slot
        const int slot = idx[b];
        const float* cs = conv_state + ((size_t)slot * Hs + h) * Km1;
        float*       os = out_state  + ((size_t)slot * Hs + h) * Km1;
        float conv = bx[r] * cw[Km1];
        float prev[8];
        for (int t = 0; t < Km1; ++t) { prev[t] = cs[t]; conv += prev[t] * cw[t]; }
        for (int t = 0; t + 1 < Km1; ++t) os[t] = prev[t + 1];  // shift window
        os[Km1 - 1] = bx[r];                                    // newest tap
        z[(size_t)b * Hs + h] = accC[r] * conv;
    }
}

// ---------------------------------------------------------------------------
// Kernel 3: y = z @ w_out.T. Block = 8 waves sharing one LDS-staged A tile;
// each wave computes a 16x32 output strip (2 adjacent 16x16 tiles) so one
// A fragment feeds two WMMAs (1 LDS + 2 global loads per 2 WMMAs).
// ---------------------------------------------------------------------------
__global__ void __launch_bounds__(256)
lfm2_out_proj_kernel(const float* __restrict__ z,      // [B,H]
                     const float* __restrict__ w_out,  // [H,H]
                     float* __restrict__ y,            // [B,H]
                     int B, int H) {
    __shared__ float As[16 * LDA];

    const int tid  = threadIdx.x;
    const int lane = tid & 31;
    const int wave = tid >> 5;
    const int hi16 = lane >> 4;
    const int l15  = lane & 15;
    const int i  = blockIdx.y;
    const int j0 = blockIdx.x * 16 + wave * 2;   // first of two 16-col tiles
    const size_t Hs = (size_t)H;

    const int arow = tid >> 4;
    const int acol = (tid & 15) << 2;
    const float* Aglb = z + (size_t)(i * 16 + arow) * Hs + acol;
    float4* Alds = (float4*)&As[arow * LDA + acol];

    const float* Bp0 = w_out + (size_t)((j0 + 0) * 16 + l15) * Hs + 2 * hi16;
    const float* Bp1 = w_out + (size_t)((j0 + 1) * 16 + l15) * Hs + 2 * hi16;

    v8f acc0 = {}, acc1 = {};
    for (int k0 = 0; k0 < H; k0 += KC) {
        __syncthreads();
        *Alds = *(const float4*)(Aglb + k0);
        __builtin_prefetch(Bp0 + k0 + KC, 0, 0);
        __builtin_prefetch(Bp1 + k0 + KC, 0, 0);
        __syncthreads();
#pragma unroll
        for (int kk = 0; kk < KC; kk += 4) {
            v2f a  = *(const v2f*)&As[l15 * LDA + kk + 2 * hi16];
            v2f b0 = *(const v2f*)(Bp0 + k0 + kk);
            v2f b1 = *(const v2f*)(Bp1 + k0 + kk);
            acc0 = wmma_f32_k4(a, b0, acc0);
            acc1 = wmma_f32_k4(a, b1, acc1);
        }
    }

    const int h0 = (j0 + 0) * 16 + l15;
    const int h1 = (j0 + 1) * 16 + l15;
    for (int r = 0; r < 8; ++r) {
        const int b = i * 16 + r + 8 * hi16;
        y[(size_t)b * Hs + h0] = acc0[r];
        y[(size_t)b * Hs + h1] = acc1[r];
    }
}

extern "C" void kernel_launch(void* const* d_in, const int* in_sizes, int n_in,
                              void* d_out, int out_size, void* d_ws, size_t ws_size,
                              hipStream_t stream) {
    const float* hid        = (const float*)d_in[0];  // [B,H]
    const float* conv_state = (const float*)d_in[1];  // [P,H,K-1]
    const int*   idx        = (const int*)  d_in[2];  // [B]
    const float* w_in       = (const float*)d_in[3];  // [3H,H]
    const float* w_out      = (const float*)d_in[4];  // [H,H]
    const float* conv_w     = (const float*)d_in[5];  // [H,K]

    const int       B = in_sizes[2];                       // 2048
    const long long H = (long long)in_sizes[0] / B;        // 2048
    const int       K = (int)((long long)in_sizes[5] / H); // 4
    const size_t nstate = (size_t)in_sizes[1];             // P*H*(K-1)

    float* y         = (float*)d_out;                 // [B,H]
    float* out_state = y + (size_t)B * H;             // [P,H,K-1]
    float* z         = (float*)d_ws;                  // [B,H] scratch (16.8 MB)

    // 1) baseline state copy (overwritten for scattered rows by kernel 2)
    lfm2_copy_state_kernel<<<2048, 256, 0, stream>>>(conv_state, out_state, nstate);

    // 2) fused in_proj + gate + depthwise conv + state scatter
    {
        dim3 grid((unsigned)(H / 128), (unsigned)(B / 16));   // 8 j-tiles / block
        lfm2_proj_conv_kernel<<<grid, 256, 0, stream>>>(
            hid, w_in, conv_state, idx, conv_w, z, out_state, B, (int)H, K);
    }

    // 3) out_proj GEMM (each wave: 16x32 strip)
    {
        dim3 grid((unsigned)(H / 256), (unsigned)(B / 16));   // 16 j-tiles / block
        lfm2_out_proj_kernel<<<grid, 256, 0, stream>>>(z, w_out, y, B, (int)H);
    }
}